// RetinaNetModelPostProcessing_50208167690830
// MI455X (gfx1250) — compile-verified
//
#include <hip/hip_runtime.h>
#include <stdint.h>

// ---------------- constants ----------------
#define IMG_F       800.0f
#define C_CLS       91
#define HWA_TOTAL   120087
#define TOPK_LVL    1000
#define CAND        5000
#define DETS        300
#define DETS_PAD    304                      // padded to 19 full 16-row WMMA tiles
#define NEG_SCORE   (-1000000000.0f)
#define SCORE_T     0.05f
#define LOGIT_T     (-2.9444389791664403f)   // log(0.05/0.95)
#define BBOX_CLAMP  4.135166556742356f       // ln(1000/16)
#define NMS_T       0.5f
#define NBINS       4096
#define BIN_LO      (-8.0f)
#define BIN_SCALE   256.0f                   // 4096 bins over [-8,8]
#define CAPC        4096
#define NMSW        157                      // ceil(5000/32)
#define TOT_ELEMS   21855834                 // 2 * 120087 * 91
#define CHUNK       1024                     // elems per block-chunk (256 thr * 4)
#define NCHUNKS     ((TOT_ELEMS + CHUNK - 1) / CHUNK)
#define STREAM_BLOCKS 1344

typedef __attribute__((ext_vector_type(2))) float v2f;
typedef __attribute__((ext_vector_type(8))) float v8f;
typedef unsigned long long u64;

__constant__ int   c_SB[11]   = {0, 8190000, 10237500, 10749375, 10887786, 10927917,
                                 19117917, 21165417, 21677292, 21815703, 21855834};
__constant__ int   c_AOFF[6]  = {0, 90000, 112500, 118125, 119646, 120087};
__constant__ int   c_W[5]     = {100, 50, 25, 13, 7};
__constant__ float c_STRIDE[5]= {8.f, 16.f, 32.f, 61.f, 114.f};
__constant__ float c_SIZE[5]  = {32.f, 64.f, 128.f, 256.f, 512.f};

// ---------------- CDNA5 async-to-LDS helpers ----------------
__device__ __forceinline__ void async_copy_b128(unsigned lds_byte_off, const void* gptr) {
  asm volatile("global_load_async_to_lds_b128 %0, %1, off"
               :: "v"(lds_byte_off), "v"(gptr) : "memory");
}
__device__ __forceinline__ void wait_async0() { asm volatile("s_wait_asynccnt 0x0" ::: "memory"); }
__device__ __forceinline__ void wait_async1() { asm volatile("s_wait_asynccnt 0x1" ::: "memory"); }

// monotonic uint key for float ordering
__device__ __forceinline__ unsigned mkey_f(float x) {
  unsigned u = __float_as_uint(x);
  return (u & 0x80000000u) ? ~u : (u | 0x80000000u);
}
__device__ __forceinline__ float unmkey_f(unsigned m) {
  unsigned u = (m & 0x80000000u) ? (m & 0x7FFFFFFFu) : ~m;
  return __uint_as_float(u);
}

// ---------------- init ----------------
__global__ void init_kernel(unsigned* hist, unsigned* cnt) {
  int i = blockIdx.x * blockDim.x + threadIdx.x;
  if (i < 10 * NBINS) hist[i] = 0u;
  if (i < 16) cnt[i] = 0u;
}

// ---------------- anchors ----------------
__global__ void anchors_kernel(float* anchors) {
  int a = blockIdx.x * blockDim.x + threadIdx.x;
  if (a >= HWA_TOTAL) return;
  int lvl = 0;
  while (lvl < 4 && a >= c_AOFF[lvl + 1]) ++lvl;
  int local = a - c_AOFF[lvl];
  int cell = local / 9, aa = local % 9;
  int W = c_W[lvl];
  int y = cell / W, x = cell % W;
  int i = aa % 3, j = aa / 3;                 // scale idx, aspect idx
  float scale = c_SIZE[lvl] * exp2f((float)i / 3.0f);
  float ar = (j == 0) ? 0.5f : ((j == 1) ? 1.0f : 2.0f);
  float hr = sqrtf(ar), wr = 1.0f / hr;
  float ws = wr * scale, hs = hr * scale;
  float b0 = rintf(-ws * 0.5f), b1 = rintf(-hs * 0.5f);
  float b2 = rintf( ws * 0.5f), b3 = rintf( hs * 0.5f);
  float sx = (float)x * c_STRIDE[lvl], sy = (float)y * c_STRIDE[lvl];
  anchors[a * 4 + 0] = sx + b0;
  anchors[a * 4 + 1] = sy + b1;
  anchors[a * 4 + 2] = sx + b2;
  anchors[a * 4 + 3] = sy + b3;
}

// ---------------- streaming scan (hist / compact) via async-to-LDS ----------------
template <int MODE>   // 0 = histogram, 1 = compact
__global__ void stream_scan(const float* __restrict__ logits,
                            unsigned* __restrict__ hist,
                            const int* __restrict__ thrBin,
                            u64* __restrict__ compact,
                            unsigned* __restrict__ cnt) {
  __shared__ float buf[2][CHUNK];
  __shared__ unsigned lhist[(MODE == 0) ? NBINS : 1];
  const int tid = threadIdx.x;                       // 256 threads
  const int cpb = (NCHUNKS + gridDim.x - 1) / gridDim.x;
  const int c0 = blockIdx.x * cpb;
  const int c1 = (c0 + cpb < NCHUNKS) ? (c0 + cpb) : NCHUNKS;
  if (c0 >= c1) return;

  int tag = 0;
  if (MODE == 0) {
    int e0 = c0 * CHUNK;
    while (tag < 9 && e0 >= c_SB[tag + 1]) ++tag;
    for (int i = tid; i < NBINS; i += 256) lhist[i] = 0u;
    __syncthreads();
  }

  auto issue = [&](int c, int b) -> bool {
    int base = c * CHUNK + tid * 4;
    if (base + 3 < TOT_ELEMS) {
      unsigned lo = (unsigned)(uintptr_t)(&buf[b][tid * 4]);
      async_copy_b128(lo, (const void*)(logits + base));
      return true;
    }
    return false;
  };

  int b = 0;
  issue(c0, 0);
  for (int c = c0; c < c1; ++c) {
    bool hasNext = (c + 1) < c1;
    bool issuedNext = false;
    if (hasNext) issuedNext = issue(c + 1, b ^ 1);
    // wait until THIS thread's chunk-c load has completed (loads retire in order)
    if (issuedNext) wait_async1(); else wait_async0();

    int base = c * CHUNK + tid * 4;
    bool tail = (base + 3 >= TOT_ELEMS);
    float v0 = 0.f, v1 = 0.f, v2 = 0.f, v3 = 0.f;
    if (!tail) {
      v0 = buf[b][tid * 4 + 0]; v1 = buf[b][tid * 4 + 1];
      v2 = buf[b][tid * 4 + 2]; v3 = buf[b][tid * 4 + 3];
    }
    #pragma unroll
    for (int k = 0; k < 4; ++k) {
      int e = base + k;
      if (e < TOT_ELEMS) {
        float x;
        if (tail) x = logits[e];
        else      x = (k == 0) ? v0 : (k == 1) ? v1 : (k == 2) ? v2 : v3;
        if (x > LOGIT_T) {
          int bin = (int)((x - BIN_LO) * BIN_SCALE);
          bin = bin < 0 ? 0 : (bin > NBINS - 1 ? NBINS - 1 : bin);
          int s = 0;
          while (s < 9 && e >= c_SB[s + 1]) ++s;
          if (MODE == 0) {
            if (s == tag) atomicAdd(&lhist[bin], 1u);
            else          atomicAdd(&hist[s * NBINS + bin], 1u);
          } else {
            if (bin >= thrBin[s]) {
              unsigned idx = (unsigned)(e - c_SB[s]);
              unsigned pos = atomicAdd(&cnt[s], 1u);
              if (pos < CAPC)
                compact[s * CAPC + pos] =
                    ((u64)mkey_f(x) << 32) | (u64)(0xFFFFFFFFu - idx);
            }
          }
        }
      }
    }
    b ^= 1;
  }
  if (MODE == 0) {
    __syncthreads();
    for (int i = tid; i < NBINS; i += 256) {
      unsigned h = lhist[i];
      if (h) atomicAdd(&hist[tag * NBINS + i], h);
    }
  }
}

// ---------------- per-stream threshold bin ----------------
__global__ void thresh_kernel(const unsigned* __restrict__ hist, int* thrBin) {
  int s = threadIdx.x;
  if (s >= 10) return;
  unsigned acc = 0; int thr = 0;
  for (int b = NBINS - 1; b >= 0; --b) {
    acc += hist[s * NBINS + b];
    if (acc >= (unsigned)TOPK_LVL) { thr = b; break; }
  }
  thrBin[s] = thr;
}

// ---------------- bitonic sort (descending, unique u64 keys) ----------------
__device__ __forceinline__ void bitonic_sort_desc(u64* a, int n, int tid, int nthr) {
  for (int k = 2; k <= n; k <<= 1) {
    for (int j = k >> 1; j > 0; j >>= 1) {
      __syncthreads();
      for (int i = tid; i < n; i += nthr) {
        int ixj = i ^ j;
        if (ixj > i) {
          u64 x = a[i], y = a[ixj];
          bool up = ((i & k) == 0);   // descending region
          if ((x < y) == up) { a[i] = y; a[ixj] = x; }
        }
      }
    }
  }
  __syncthreads();
}

// ---------------- per-stream top-1000 + decode ----------------
__global__ void select_kernel(const u64* __restrict__ compact,
                              const unsigned* __restrict__ cnt,
                              const float* __restrict__ reg,
                              const float* __restrict__ anchors,
                              float* candBox, float* candScore, int* candLabel) {
  __shared__ u64 arr[CAPC];
  int s = blockIdx.x, tid = threadIdx.x;
  int img = s / 5, lvl = s % 5;
  unsigned cn = cnt[s];
  int n = (cn < (unsigned)CAPC) ? (int)cn : CAPC;
  for (int i = tid; i < CAPC; i += blockDim.x)
    arr[i] = (i < n) ? compact[s * CAPC + i] : 0ull;
  __syncthreads();
  bitonic_sort_desc(arr, CAPC, tid, blockDim.x);

  if (tid < TOPK_LVL) {
    int slot = img * CAND + lvl * TOPK_LVL + tid;
    float score, b0 = 0.f, b1 = 0.f, b2 = 0.f, b3 = 0.f;
    int lab = 0;
    if (tid < n) {
      u64 v = arr[tid];
      unsigned mk = (unsigned)(v >> 32);
      unsigned idx = 0xFFFFFFFFu - (unsigned)v;
      float x = unmkey_f(mk);
      score = 1.0f / (1.0f + expf(-x));
      lab = (int)(idx % (unsigned)C_CLS);
      int aloc = (int)(idx / (unsigned)C_CLS);
      int aidx = c_AOFF[lvl] + aloc;
      const float* an = anchors + (size_t)aidx * 4;
      const float* rg = reg + ((size_t)img * HWA_TOTAL + aidx) * 4;
      float wa = an[2] - an[0], ha = an[3] - an[1];
      float cxa = an[0] + 0.5f * wa, cya = an[1] + 0.5f * ha;
      float dx = rg[0], dy = rg[1];
      float dw = fminf(rg[2], BBOX_CLAMP), dh = fminf(rg[3], BBOX_CLAMP);
      float cx = dx * wa + cxa, cy = dy * ha + cya;
      float w = expf(dw) * wa, h = expf(dh) * ha;
      b0 = fminf(fmaxf(cx - 0.5f * w, 0.f), IMG_F);
      b1 = fminf(fmaxf(cy - 0.5f * h, 0.f), IMG_F);
      b2 = fminf(fmaxf(cx + 0.5f * w, 0.f), IMG_F);
      b3 = fminf(fmaxf(cy + 0.5f * h, 0.f), IMG_F);
    } else {
      score = NEG_SCORE;
    }
    candScore[slot] = score;
    candLabel[slot] = lab;
    candBox[slot * 4 + 0] = b0; candBox[slot * 4 + 1] = b1;
    candBox[slot * 4 + 2] = b2; candBox[slot * 4 + 3] = b3;
  }
}

// ---------------- per-image sort of 5000 candidates ----------------
__global__ void sortcand_kernel(const float* __restrict__ candBox,
                                const float* __restrict__ candScore,
                                const int* __restrict__ candLabel,
                                float* sBox, float* sScore, int* sLabel, float* offBox) {
  __shared__ u64 arr[8192];
  int img = blockIdx.x, tid = threadIdx.x;
  for (int i = tid; i < 8192; i += blockDim.x) {
    u64 v = 0ull;
    if (i < CAND) {
      float sc = candScore[img * CAND + i];
      v = ((u64)mkey_f(sc) << 32) | (u64)(0xFFFFFFFFu - (unsigned)i);
    }
    arr[i] = v;
  }
  __syncthreads();
  bitonic_sort_desc(arr, 8192, tid, blockDim.x);
  for (int j = tid; j < CAND; j += blockDim.x) {
    int src = (int)(0xFFFFFFFFu - (unsigned)arr[j]);
    int sslot = img * CAND + src, dslot = img * CAND + j;
    float b0 = candBox[sslot * 4 + 0], b1 = candBox[sslot * 4 + 1];
    float b2 = candBox[sslot * 4 + 2], b3 = candBox[sslot * 4 + 3];
    int lab = candLabel[sslot];
    sBox[dslot * 4 + 0] = b0; sBox[dslot * 4 + 1] = b1;
    sBox[dslot * 4 + 2] = b2; sBox[dslot * 4 + 3] = b3;
    sScore[dslot] = candScore[sslot];
    sLabel[dslot] = lab;
    float off = (float)lab * (IMG_F + 1.0f);
    offBox[dslot * 4 + 0] = b0 + off; offBox[dslot * 4 + 1] = b1 + off;
    offBox[dslot * 4 + 2] = b2 + off; offBox[dslot * 4 + 3] = b3 + off;
  }
}

// ---------------- NMS suppression bitmask ----------------
__global__ void nmsmask_kernel(const float* __restrict__ offBox, unsigned* __restrict__ mask) {
  int idx = blockIdx.x * blockDim.x + threadIdx.x;
  const int TOTM = 2 * CAND * NMSW;
  if (idx >= TOTM) return;
  int img = idx / (CAND * NMSW);
  int r = idx % (CAND * NMSW);
  int i = r / NMSW, w = r % NMSW;
  const float4* B = (const float4*)(offBox + (size_t)img * CAND * 4);
  float4 bi = B[i];
  float ai = (bi.z - bi.x) * (bi.w - bi.y);
  unsigned bits = 0u;
  int j0 = w * 32;
  #pragma unroll 4
  for (int k = 0; k < 32; ++k) {
    int j = j0 + k;
    if (j < CAND) {
      float4 bj = B[j];
      float aj = (bj.z - bj.x) * (bj.w - bj.y);
      float lx = fmaxf(bi.x, bj.x), ly = fmaxf(bi.y, bj.y);
      float rx = fminf(bi.z, bj.z), ry = fminf(bi.w, bj.w);
      float iw = fmaxf(rx - lx, 0.f), ih = fmaxf(ry - ly, 0.f);
      float inter = iw * ih;
      float iou = inter / (ai + aj - inter + 1e-7f);
      if (iou > NMS_T) bits |= (1u << k);
    }
  }
  mask[idx] = bits;
}

// ---------------- greedy reduce + outputs + WMMA rescale epilogue ----------------
__global__ void finalize_kernel(const float* __restrict__ sBox,
                                const float* __restrict__ sScore,
                                const int* __restrict__ sLabel,
                                const unsigned* __restrict__ mask,
                                float* __restrict__ out) {
  __shared__ unsigned remv[NMSW];
  __shared__ float svals[CAND];
  __shared__ int keptList[DETS];
  __shared__ float obox[DETS_PAD * 4];        // zero-padded to 19 full tiles
  int img = blockIdx.x, lane = threadIdx.x;   // 32 threads = 1 wave

  for (int i = lane; i < NMSW; i += 32) remv[i] = 0u;
  for (int i = lane; i < CAND; i += 32) svals[i] = sScore[img * CAND + i];
  __syncthreads();   // single-wave workgroup -> S_NOP

  int kept = 0;
  for (int i = 0; i < CAND && kept < DETS; ++i) {
    unsigned wv = remv[i >> 5];
    if (!((wv >> (i & 31)) & 1u) && svals[i] > SCORE_T) {
      if (lane == 0) keptList[kept] = i;
      const unsigned* mrow = mask + (size_t)img * CAND * NMSW + (size_t)i * NMSW;
      for (int wq = lane; wq < NMSW; wq += 32) remv[wq] |= mrow[wq];
      ++kept;
    }
  }
  __syncthreads();

  // scores, labels to global; boxes stashed (zero-padded) in LDS for the WMMA rescale
  for (int t = lane; t < DETS_PAD; t += 32) {
    float sc = 0.f, lb = -1.f;
    float b0 = 0.f, b1 = 0.f, b2 = 0.f, b3 = 0.f;
    if (t < kept) {
      int i = keptList[t];
      sc = svals[i];
      lb = (float)sLabel[img * CAND + i];
      const float* bp = sBox + ((size_t)img * CAND + i) * 4;
      b0 = bp[0]; b1 = bp[1]; b2 = bp[2]; b3 = bp[3];
    }
    if (t < DETS) {
      out[2 * DETS * 4 + img * DETS + t] = sc;            // scores at 2400..
      out[2 * DETS * 4 + 2 * DETS + img * DETS + t] = lb; // labels at 3000..
    }
    obox[t * 4 + 0] = b0; obox[t * 4 + 1] = b1;
    obox[t * 4 + 2] = b2; obox[t * 4 + 3] = b3;
  }
  __syncthreads();

  // boxes_out(300x4) = obox(300x4) x diag(rw,rh,rw,rh) via V_WMMA_F32_16X16X4_F32
  const float rw = 1.0f, rh = 1.0f;   // image is 800x800 -> ratio 1
  float scl0 = rw, scl1 = rh, scl2 = rw, scl3 = rh;
  int half = lane >> 4;          // 0: K=0,1 | 1: K=2,3
  int nn = lane & 15;
  v2f bmat;
  {
    int k0 = half * 2, k1 = half * 2 + 1;
    float s0 = (k0 == 0) ? scl0 : (k0 == 1) ? scl1 : (k0 == 2) ? scl2 : scl3;
    float s1 = (k1 == 0) ? scl0 : (k1 == 1) ? scl1 : (k1 == 2) ? scl2 : scl3;
    bmat.x = (nn == k0) ? s0 : 0.f;
    bmat.y = (nn == k1) ? s1 : 0.f;
  }
  for (int tile = 0; tile < DETS_PAD / 16; ++tile) {
    int rrow = tile * 16 + (lane & 15);          // < DETS_PAD, always in range
    v2f amat;
    amat.x = obox[rrow * 4 + half * 2 + 0];
    amat.y = obox[rrow * 4 + half * 2 + 1];
    v8f cz = {0.f, 0.f, 0.f, 0.f, 0.f, 0.f, 0.f, 0.f};
    v8f d = __builtin_amdgcn_wmma_f32_16x16x4_f32(
        false, amat, false, bmat, (short)0, cz, false, false);
    if (nn < 4) {
      #pragma unroll
      for (int r8 = 0; r8 < 8; ++r8) {
        int row = tile * 16 + r8 + half * 8;
        if (row < DETS) out[((size_t)img * DETS + row) * 4 + nn] = d[r8];
      }
    }
  }
}

// ---------------- host launch ----------------
extern "C" void kernel_launch(void* const* d_in, const int* in_sizes, int n_in,
                              void* d_out, int out_size, void* d_ws, size_t ws_size,
                              hipStream_t stream) {
  (void)in_sizes; (void)n_in; (void)out_size; (void)ws_size;
  const float* logits = (const float*)d_in[6];
  const float* reg    = (const float*)d_in[7];
  float* out = (float*)d_out;

  char* w = (char*)d_ws;
  size_t off = 0;
  auto alloc = [&](size_t bytes) -> void* {
    void* p = (void*)(w + off);
    off = (off + bytes + 255) & ~(size_t)255;
    return p;
  };
  float*    anchors   = (float*)alloc((size_t)HWA_TOTAL * 4 * 4);
  unsigned* hist      = (unsigned*)alloc((size_t)10 * NBINS * 4);
  int*      thrBin    = (int*)alloc(64);
  unsigned* cnt       = (unsigned*)alloc(64);
  u64*      compact   = (u64*)alloc((size_t)10 * CAPC * 8);
  float*    candBox   = (float*)alloc((size_t)2 * CAND * 4 * 4);
  float*    candScore = (float*)alloc((size_t)2 * CAND * 4);
  int*      candLabel = (int*)alloc((size_t)2 * CAND * 4);
  float*    sBox      = (float*)alloc((size_t)2 * CAND * 4 * 4);
  float*    sScore    = (float*)alloc((size_t)2 * CAND * 4);
  int*      sLabel    = (int*)alloc((size_t)2 * CAND * 4);
  float*    offBox    = (float*)alloc((size_t)2 * CAND * 4 * 4);
  unsigned* mask      = (unsigned*)alloc((size_t)2 * CAND * NMSW * 4);

  init_kernel<<<(10 * NBINS + 255) / 256, 256, 0, stream>>>(hist, cnt);
  anchors_kernel<<<(HWA_TOTAL + 255) / 256, 256, 0, stream>>>(anchors);
  stream_scan<0><<<STREAM_BLOCKS, 256, 0, stream>>>(logits, hist, nullptr, nullptr, nullptr);
  thresh_kernel<<<1, 32, 0, stream>>>(hist, thrBin);
  stream_scan<1><<<STREAM_BLOCKS, 256, 0, stream>>>(logits, hist, thrBin, compact, cnt);
  select_kernel<<<10, 1024, 0, stream>>>(compact, cnt, reg, anchors,
                                         candBox, candScore, candLabel);
  sortcand_kernel<<<2, 1024, 0, stream>>>(candBox, candScore, candLabel,
                                          sBox, sScore, sLabel, offBox);
  nmsmask_kernel<<<(2 * CAND * NMSW + 255) / 256, 256, 0, stream>>>(offBox, mask);
  finalize_kernel<<<2, 32, 0, stream>>>(sBox, sScore, sLabel, mask, out);
}